// BiquadCellWithSidechain_1597727834564
// MI455X (gfx1250) — compile-verified
//
#include <hip/hip_runtime.h>
#include <hip/hip_bf16.h>

// ---------------------------------------------------------------------------
// Kernel 1: parallel precompute of per-step packed coefficients.
//   pack[t*B + b] = { ff, a, c, 0 }
//   ff = sum_i x[t,b,i]*(w[i]+sc[t,b,i]);  a = w[3]+sc[t,b,3];  c = w[4]+sc[t,b,4]
// Pure bandwidth (~134 MB read NT, 67 MB write RT so it stays in the 192MB L2
// for the scan kernel) -> ~9 us at 23.3 TB/s.
// ---------------------------------------------------------------------------
__global__ __launch_bounds__(256) void biquad_precompute(
    const float* __restrict__ x, const float* __restrict__ sc,
    const float* __restrict__ w, float4* __restrict__ pack, int total)
{
    int i = blockIdx.x * blockDim.x + threadIdx.x;   // i = t*B + b
    if (i >= total) return;
    const float* xi = x  + (size_t)i * 3;
    const float* si = sc + (size_t)i * 5;
    float w0 = w[0], w1 = w[1], w2 = w[2], w3 = w[3], w4 = w[4];
    float x0 = __builtin_nontemporal_load(xi + 0);
    float x1 = __builtin_nontemporal_load(xi + 1);
    float x2 = __builtin_nontemporal_load(xi + 2);
    float s0 = __builtin_nontemporal_load(si + 0);
    float s1 = __builtin_nontemporal_load(si + 1);
    float s2 = __builtin_nontemporal_load(si + 2);
    float s3 = __builtin_nontemporal_load(si + 3);
    float s4 = __builtin_nontemporal_load(si + 4);
    float ff = x0 * (w0 + s0);
    ff = fmaf(x1, (w1 + s1), ff);
    ff = fmaf(x2, (w2 + s2), ff);
    pack[i] = make_float4(ff, w3 + s3, w4 + s4, 0.0f);
}

// ---------------------------------------------------------------------------
// Kernel 2: serial tanh recurrence, one chain per lane, one wave32 per block.
// 64-slot LDS ring filled by async global->LDS b128 DMAs (ASYNCcnt), consumed
// via software-pipelined ds_load_b128 one step ahead, so the per-step critical
// path is only 2x v_fmac + v_tanh_f32.
//
// LDS addresses are produced by ptrtoint of a real AS(3) pointer into smem so
// the address escapes into the asm: this both yields the correct LDS byte
// offset and prevents LLVM from proving smem "never written" and deleting the
// ds_load consumers (which happened when passing hand-computed integers).
// ---------------------------------------------------------------------------
#define RING_D 64   // must be 64: s_wait_asynccnt 63 is then the exact guard

typedef __attribute__((address_space(3))) char lds_char_t;

__global__ __launch_bounds__(32) void biquad_scan_async(
    const float4* __restrict__ pack, const float* __restrict__ carry0,
    float* __restrict__ out, int T, int B)
{
    __shared__ float4 smem[RING_D * 32];             // 32 KB ring buffer

    const int lane = threadIdx.x;                    // 0..31
    const int b    = blockIdx.x * 32 + lane;         // chain index (B % 32 == 0)

    const unsigned long long base = (unsigned long long)pack;

    // Real LDS byte address of this lane's slot-0 element (escapes smem).
    const unsigned lds0 =
        (unsigned)(size_t)(lds_char_t*)&smem[lane];
    const unsigned lds_slot_stride = 32u * sizeof(float4);   // bytes per slot

    // Prologue: fill the ring RING_D steps deep (requires T >= RING_D).
    for (int s = 0; s < RING_D; ++s) {
        unsigned lds  = lds0 + (unsigned)s * lds_slot_stride;
        unsigned goff = (unsigned)(s * B + b) * 16u;
        asm volatile("global_load_async_to_lds_b128 %0, %1, %2"
                     :: "v"(lds), "v"(goff), "s"(base) : "memory");
    }

    float o1 = carry0[2 * b + 0];   // o[t-1]
    float o2 = carry0[2 * b + 1];   // o[t-2]

    // Load slot 0 into registers.
    asm volatile("s_wait_asynccnt 63" ::: "memory");
    float4 cur = smem[lane];
    asm volatile("s_wait_dscnt 0" ::: "memory");

    for (int t = 0; t < T; ++t) {
        const int slot = t & (RING_D - 1);

        // Refill the slot we just consumed (cur is already in VGPRs, dscnt==0,
        // so no LDS WAR hazard) with data for step t + RING_D.
        int tp = t + RING_D;
        if (tp < T) {
            unsigned lds  = lds0 + (unsigned)slot * lds_slot_stride;
            unsigned goff = (unsigned)(tp * B + b) * 16u;
            asm volatile("global_load_async_to_lds_b128 %0, %1, %2"
                         :: "v"(lds), "v"(goff), "s"(base) : "memory");
        }

        // Prefetch next step's coefficients from LDS; waited only at loop end,
        // so the ds latency overlaps the tanh chain below.
        float4 nxt = cur;
        if (t + 1 < T) {
            // After the issue above: issued = t + 65; need load #(t+2) done
            // => asynccnt <= 63.  In-order completion makes this exact.
            asm volatile("s_wait_asynccnt 63" ::: "memory");
            nxt = smem[((t + 1) & (RING_D - 1)) * 32 + lane];
        }

        // acc = ff + a*o[t-1] + c*o[t-2]
        float acc = fmaf(cur.y, o1, fmaf(cur.z, o2, cur.x));

        // Native gfx1250 tanh; v_nop satisfies the TRANS-op result hazard
        // (inline asm bypasses compiler hazard insertion).
        float o;
        asm volatile("v_tanh_f32 %0, %1\n\tv_nop"
                     : "=v"(o) : "v"(acc));

        __builtin_nontemporal_store(o, &out[(size_t)t * B + b]);
        o2 = o1;
        o1 = o;

        asm volatile("s_wait_dscnt 0" ::: "memory");   // nxt now in VGPRs
        cur = nxt;
    }
}

// ---------------------------------------------------------------------------
// Fallback serial kernel (workspace too small / odd shapes): direct loads.
// ---------------------------------------------------------------------------
__global__ __launch_bounds__(64) void biquad_scan_direct(
    const float* __restrict__ x, const float* __restrict__ sc,
    const float* __restrict__ carry0, const float* __restrict__ w,
    float* __restrict__ out, int T, int B)
{
    int b = blockIdx.x * blockDim.x + threadIdx.x;
    if (b >= B) return;
    float w0 = w[0], w1 = w[1], w2 = w[2], w3 = w[3], w4 = w[4];
    float o1 = carry0[2 * b + 0];
    float o2 = carry0[2 * b + 1];
    for (int t = 0; t < T; ++t) {
        const float* xi = x  + ((size_t)t * B + b) * 3;
        const float* si = sc + ((size_t)t * B + b) * 5;
        float acc = xi[0] * (w0 + si[0]);
        acc = fmaf(xi[1], (w1 + si[1]), acc);
        acc = fmaf(xi[2], (w2 + si[2]), acc);
        acc = fmaf(o1,    (w3 + si[3]), acc);
        acc = fmaf(o2,    (w4 + si[4]), acc);
        float o;
        asm volatile("v_tanh_f32 %0, %1\n\tv_nop" : "=v"(o) : "v"(acc));
        out[(size_t)t * B + b] = o;
        o2 = o1;
        o1 = o;
    }
}

// ---------------------------------------------------------------------------
extern "C" void kernel_launch(void* const* d_in, const int* in_sizes, int n_in,
                              void* d_out, int out_size, void* d_ws, size_t ws_size,
                              hipStream_t stream) {
    const float* x  = (const float*)d_in[0];   // (T, B, 3)
    const float* sc = (const float*)d_in[1];   // (T, B, 5)
    const float* c0 = (const float*)d_in[2];   // (B, 2)
    const float* w  = (const float*)d_in[3];   // (1, 5)
    float* out = (float*)d_out;                // (T, B, 1)

    const int B = in_sizes[2] / 2;
    const int T = in_sizes[0] / (3 * B);
    const int total = T * B;

    const size_t need = (size_t)total * sizeof(float4);

    if (ws_size >= need && (B % 32) == 0 && T >= RING_D) {
        float4* pack = (float4*)d_ws;
        biquad_precompute<<<(total + 255) / 256, 256, 0, stream>>>(x, sc, w, pack, total);
        biquad_scan_async<<<B / 32, 32, 0, stream>>>(pack, c0, out, T, B);
    } else {
        biquad_scan_direct<<<(B + 63) / 64, 64, 0, stream>>>(x, sc, c0, w, out, T, B);
    }
}